// ModelToOnnx_22299470201307
// MI455X (gfx1250) — compile-verified
//
#include <hip/hip_runtime.h>
#include <hip/hip_bf16.h>
#include <stdint.h>

typedef __attribute__((ext_vector_type(2))) float v2f;
typedef __attribute__((ext_vector_type(8))) float v8f;

#define NPRI   172032
#define NBLK   672          // 672*256 == 172032 exactly
#define TOPK   4096
#define CONF_T 0.7f
#define NMS_T  0.4f

// ---- workspace layout (bytes) ----
#define KEYS_OFF   0u          // u32 * 172032           -> 688128
#define HIST_OFF   688128u     // u32 * 1024 (4 passes)  -> 692224
#define STATE_OFF  692224u     // u32 * 16               -> 692288
#define CAND_OFF   692288u     // u64 * 4096             -> 725056
#define BOX_OFF    725056u     // f32 * 4096*4           -> 790592
#define LM_OFF     790592u     // f32 * 4096*10          -> 954432
#define SC_OFF     954432u     // f32 * 4096             -> 970816
#define MASK_OFF   970816u     // u16 * 4096*256         -> 3067968
#define KEEP_OFF   3067968u    // u32 * 4096             -> 3084352

// state[0]=prefix/threshold  state[1]=count strictly above  state[2]=hi slot ctr  state[3]=eq ctr

__global__ void k_zero(uint32_t* hist, uint32_t* state) {
  int t = threadIdx.x;
  for (int i = t; i < 1024; i += 256) hist[i] = 0u;
  if (t < 16) state[t] = 0u;
}

// score = softmax(conf)[:,1] = sigmoid(c1-c0); masked; order-preserving u32 key (bigger = better)
__global__ void k_scorekeys(const float* __restrict__ conf, uint32_t* __restrict__ keys) {
  int p = blockIdx.x * 256 + threadIdx.x;
  float c0 = conf[2 * p], c1 = conf[2 * p + 1];
  float s = 1.0f / (1.0f + __expf(c0 - c1));
  float m = (s > CONF_T) ? s : -1.0f;
  uint32_t u = __float_as_uint(m);
  uint32_t k = (u & 0x80000000u) ? ~u : (u | 0x80000000u);
  keys[p] = k;
}

// MSD radix-select histogram, pass = 0..3 over byte (24-8*pass)
__global__ void k_hist(const uint32_t* __restrict__ keys, const uint32_t* __restrict__ state,
                       uint32_t* __restrict__ hist, int pass) {
  __shared__ uint32_t h[256];
  int t = threadIdx.x;
  h[t] = 0u;
  __syncthreads();
  int p = blockIdx.x * 256 + t;
  uint32_t k = keys[p];
  int shift = 24 - 8 * pass;
  bool ok = true;
  if (pass > 0) {
    int hs = shift + 8;
    ok = (k >> hs) == (state[0] >> hs);
  }
  if (ok) atomicAdd(&h[(k >> shift) & 0xFFu], 1u);
  __syncthreads();
  if (h[t]) atomicAdd(&hist[pass * 256 + t], h[t]);
}

__global__ void k_select(const uint32_t* __restrict__ hist, uint32_t* __restrict__ state, int pass) {
  if (threadIdx.x != 0) return;
  int shift = 24 - 8 * pass;
  uint32_t running = state[1];
  uint32_t pref = state[0];
  for (int b = 255; b >= 0; b--) {
    uint32_t c = hist[pass * 256 + b];
    if (running + c >= (uint32_t)TOPK) {
      state[0] = pref | ((uint32_t)b << shift);
      state[1] = running;
      return;
    }
    running += c;
  }
  state[1] = running;  // unreachable fallback
}

// exactly TOPK winners: all keys > T, plus first (TOPK-above) keys == T
__global__ void k_compact(const uint32_t* __restrict__ keys, uint32_t* __restrict__ state,
                          unsigned long long* __restrict__ cand) {
  int p = blockIdx.x * 256 + threadIdx.x;
  uint32_t k = keys[p];
  uint32_t T = state[0];
  if (k > T) {
    uint32_t slot = atomicAdd(&state[2], 1u);
    cand[slot] = (((unsigned long long)(~k)) << 32) | (uint32_t)p;
  } else if (k == T) {
    uint32_t ca = state[1];
    uint32_t e = atomicAdd(&state[3], 1u);
    if (e < (uint32_t)TOPK - ca)
      cand[ca + e] = (((unsigned long long)(~k)) << 32) | (uint32_t)p;
  }
}

// in-LDS bitonic sort of 4096 u64 keys (ascending key64 == descending score, ties by index)
__global__ __launch_bounds__(1024) void k_sort(unsigned long long* __restrict__ cand) {
  __shared__ unsigned long long sk[TOPK];
  int t = threadIdx.x;
  for (int i = t; i < TOPK; i += 1024) sk[i] = cand[i];
  __syncthreads();
  for (int k = 2; k <= TOPK; k <<= 1) {
    for (int j = k >> 1; j > 0; j >>= 1) {
      for (int i = t; i < TOPK; i += 1024) {
        int ixj = i ^ j;
        if (ixj > i) {
          unsigned long long a = sk[i], b = sk[ixj];
          bool up = (i & k) == 0;
          if ((a > b) == up) { sk[i] = b; sk[ixj] = a; }
        }
      }
      __syncthreads();
    }
  }
  for (int i = t; i < TOPK; i += 1024) cand[i] = sk[i];
}

// decode only the 4096 winners; prior recomputed analytically from flat index
__global__ void k_decode(const unsigned long long* __restrict__ cand,
                         const float* __restrict__ loc, const float* __restrict__ land,
                         float* __restrict__ BOX, float* __restrict__ LM, float* __restrict__ SC) {
  int t = blockIdx.x * 256 + threadIdx.x;
  unsigned long long key = cand[t];
  uint32_t idx = (uint32_t)key;
  uint32_t kk = ~(uint32_t)(key >> 32);
  uint32_t u = (kk & 0x80000000u) ? (kk & 0x7FFFFFFFu) : ~kk;  // inverse order transform
  SC[t] = __uint_as_float(u);

  int i2 = (int)idx, fw, step; float m;
  if (i2 < 131072)       { fw = 256; step = 8;  m = (i2 & 1) ? 32.f  : 16.f;  }
  else if (i2 < 163840)  { i2 -= 131072; fw = 128; step = 16; m = (i2 & 1) ? 128.f : 64.f;  }
  else                   { i2 -= 163840; fw = 64;  step = 32; m = (i2 & 1) ? 512.f : 256.f; }
  int cell = i2 >> 1;
  int y = cell / fw, x = cell - y * fw;
  const float inv = 1.0f / 2048.0f;
  float cx = (x + 0.5f) * step * inv, cy = (y + 0.5f) * step * inv;
  float sk = m * inv;  // skx == sky (W == H)

  float l0 = loc[idx * 4 + 0], l1 = loc[idx * 4 + 1];
  float l2 = loc[idx * 4 + 2], l3 = loc[idx * 4 + 3];
  float ccx = cx + l0 * 0.1f * sk;
  float ccy = cy + l1 * 0.1f * sk;
  float w = sk * __expf(l2 * 0.2f);
  float h = sk * __expf(l3 * 0.2f);
  float tlx = ccx - 0.5f * w, tly = ccy - 0.5f * h;
  BOX[t * 4 + 0] = tlx * 2048.f;
  BOX[t * 4 + 1] = tly * 2048.f;
  BOX[t * 4 + 2] = (tlx + w) * 2048.f;
  BOX[t * 4 + 3] = (tly + h) * 2048.f;
  for (int q = 0; q < 5; q++) {
    float gx = land[idx * 10 + 2 * q], gy = land[idx * 10 + 2 * q + 1];
    LM[t * 10 + 2 * q]     = (cx + gx * 0.1f * sk) * 2048.f;
    LM[t * 10 + 2 * q + 1] = (cy + gy * 0.1f * sk) * 2048.f;
  }
}

// One wave per 16x16 IoU tile. Boxes staged via async-to-LDS; pairwise area sums via
// V_WMMA_F32_16X16X4_F32 rank-2 outer product; ballot -> 16-bit mask halfwords.
__global__ __launch_bounds__(256) void k_ioumask(const float* __restrict__ BOX,
                                                 unsigned short* __restrict__ mask) {
  __shared__ __align__(16) float lds[8 * 128];  // per wave: 16 row boxes + 16 col boxes (512B)
  int wid = threadIdx.x >> 5;
  int lane = threadIdx.x & 31;
  int rt = blockIdx.y * 8 + wid;       // row tile 0..255
  int ct = blockIdx.x;                 // col tile 0..255
  int r0 = rt * 16, c0 = ct * 16;

  // lanes 0-15 fetch row boxes, lanes 16-31 fetch col boxes (16B each), async into LDS
  int src = (lane < 16) ? (r0 + lane) : (c0 + lane - 16);
  unsigned long long gaddr = (unsigned long long)(uintptr_t)(BOX + src * 4);
  float* myLds = &lds[wid * 128 + lane * 4];
  unsigned int ldsOff = (unsigned int)(uintptr_t)myLds;  // low 32 bits of generic addr = LDS offset
  asm volatile("global_load_async_to_lds_b128 %0, %1, off" :: "v"(ldsOff), "v"(gaddr) : "memory");
  asm volatile("s_wait_asynccnt 0" ::: "memory");        // issuing wave consumes its own data

  const float* wb = &lds[wid * 128];
  float4 own = *(const float4*)&wb[lane * 4];
  float areaOwn = (own.z - own.x) * (own.w - own.y);
  float areaSw = __shfl(areaOwn, lane ^ 16, 32);
  bool lo = lane < 16;
  // A(16x4): K0 = area_row, K1 = 1 ; B(4x16): K0 = 1, K1 = area_col  ->  D = area_i + area_j
  v2f Am = { lo ? areaOwn : 0.0f, lo ? 1.0f : 0.0f };
  v2f Bm = { lo ? 1.0f : 0.0f,    lo ? areaSw : 0.0f };
  v8f Cm = {0.f, 0.f, 0.f, 0.f, 0.f, 0.f, 0.f, 0.f};
  v8f S = __builtin_amdgcn_wmma_f32_16x16x4_f32(false, Am, false, Bm, (short)0, Cm, false, false);

  int n = lane & 15;                    // column within tile (matches C/D lane layout)
  float4 cb = *(const float4*)&wb[(16 + n) * 4];
  int colIdx = c0 + n;
  int half = (lane >= 16) ? 8 : 0;      // C/D: VGPR r holds row r (lanes<16) / r+8 (lanes>=16)
  for (int r = 0; r < 8; r++) {
    int M = r + half;
    float4 rb = *(const float4*)&wb[M * 4];
    float ix1 = fmaxf(rb.x, cb.x), iy1 = fmaxf(rb.y, cb.y);
    float ix2 = fminf(rb.z, cb.z), iy2 = fminf(rb.w, cb.w);
    float inter = fmaxf(ix2 - ix1, 0.f) * fmaxf(iy2 - iy1, 0.f);
    float denom = S[r] - inter + 1e-9f;            // area_i + area_j - inter + eps (WMMA result)
    bool pred = (inter > NMS_T * denom) && (colIdx > r0 + M);
    unsigned long long bal = __ballot(pred);
    if (lane == 0) {
      mask[(unsigned)(r0 + r) * 256u + ct]     = (unsigned short)(bal & 0xFFFFu);
      mask[(unsigned)(r0 + r + 8) * 256u + ct] = (unsigned short)((bal >> 16) & 0xFFFFu);
    }
  }
}

// sequential greedy NMS over the bitmask; suppression state lives in LDS (256 x u16-in-u32)
__global__ __launch_bounds__(256) void k_nms(const float* __restrict__ SC,
                                             const unsigned short* __restrict__ mask,
                                             unsigned int* __restrict__ KEEP) {
  __shared__ unsigned int sup[256];
  __shared__ int cur;
  int t = threadIdx.x;
  sup[t] = 0u;
  __syncthreads();
  for (int i = 0; i < TOPK; i++) {
    if (t == (i >> 4)) {
      bool kept = (SC[i] > CONF_T) && !((sup[t] >> (i & 15)) & 1u);
      cur = kept;
      KEEP[i] = kept ? 1u : 0u;
    }
    __syncthreads();
    if (cur) sup[t] |= (unsigned int)mask[(unsigned)i * 256u + t];
    __syncthreads();
  }
}

__global__ void k_out(const float* __restrict__ BOX, const float* __restrict__ SC,
                      const float* __restrict__ LM, const unsigned int* __restrict__ KEEP,
                      float* __restrict__ out) {
  int t = blockIdx.x * 256 + threadIdx.x;
  bool keep = KEEP[t] != 0u;
  float* ob = out;              // boxes  [4096*4]
  float* os = out + 16384;      // scores [4096]
  float* ol = out + 20480;      // lms    [4096*10]
  for (int j = 0; j < 4; j++)  ob[t * 4 + j]  = keep ? BOX[t * 4 + j]  : 0.0f;
  os[t] = keep ? SC[t] : 0.0f;
  for (int j = 0; j < 10; j++) ol[t * 10 + j] = keep ? LM[t * 10 + j] : 0.0f;
}

extern "C" void kernel_launch(void* const* d_in, const int* in_sizes, int n_in,
                              void* d_out, int out_size, void* d_ws, size_t ws_size,
                              hipStream_t stream) {
  (void)in_sizes; (void)n_in; (void)out_size; (void)ws_size;
  const float* loc  = (const float*)d_in[0];
  const float* conf = (const float*)d_in[1];
  const float* land = (const float*)d_in[2];

  uint8_t* ws = (uint8_t*)d_ws;
  uint32_t*           keys  = (uint32_t*)(ws + KEYS_OFF);
  uint32_t*           hist  = (uint32_t*)(ws + HIST_OFF);
  uint32_t*           state = (uint32_t*)(ws + STATE_OFF);
  unsigned long long* cand  = (unsigned long long*)(ws + CAND_OFF);
  float*              BOX   = (float*)(ws + BOX_OFF);
  float*              LM    = (float*)(ws + LM_OFF);
  float*              SC    = (float*)(ws + SC_OFF);
  unsigned short*     mask  = (unsigned short*)(ws + MASK_OFF);
  uint32_t*           KEEP  = (uint32_t*)(ws + KEEP_OFF);

  k_zero<<<1, 256, 0, stream>>>(hist, state);
  k_scorekeys<<<NBLK, 256, 0, stream>>>(conf, keys);
  for (int pass = 0; pass < 4; pass++) {
    k_hist<<<NBLK, 256, 0, stream>>>(keys, state, hist, pass);
    k_select<<<1, 64, 0, stream>>>(hist, state, pass);
  }
  k_compact<<<NBLK, 256, 0, stream>>>(keys, state, cand);
  k_sort<<<1, 1024, 0, stream>>>(cand);
  k_decode<<<TOPK / 256, 256, 0, stream>>>(cand, loc, land, BOX, LM, SC);
  k_ioumask<<<dim3(256, 32), 256, 0, stream>>>(BOX, mask);
  k_nms<<<1, 256, 0, stream>>>(SC, mask, KEEP);
  k_out<<<TOPK / 256, 256, 0, stream>>>(BOX, SC, LM, KEEP, (float*)d_out);
}